// BasicRecurrentEntityEncoder_58231166599769
// MI455X (gfx1250) — compile-verified
//
#include <hip/hip_runtime.h>

// Shapes (fixed by the reference)
#define B_  256
#define S_  64
#define L_  16
#define D_  256
#define K_  30
#define HS  260   // LDS row stride in floats; 260 % 64 == 4 -> conflict-free row access

typedef __attribute__((ext_vector_type(2))) float v2f;
typedef __attribute__((ext_vector_type(8))) float v8f;

// ------------------------------------------------------------------
// CDNA5 async global->LDS copy (ASYNCcnt path), per CDNA5 ISA.
// GVS addressing: mem = SADDR(64) + VADDR(i32) + IOFFSET. VDST = LDS byte addr.
// ------------------------------------------------------------------
__device__ __forceinline__ void async_ld_f32(const float* gbase, int byte_off,
                                             void* lds_ptr) {
  const unsigned lds_addr = (unsigned)(size_t)lds_ptr;   // flat LDS addr[31:0] == LDS offset
  asm volatile("global_load_async_to_lds_b32 %0, %1, %2 offset:0"
               :: "v"(lds_addr), "v"(byte_off), "s"(gbase)
               : "memory");
}
__device__ __forceinline__ void wait_async0() {
  asm volatile("s_wait_asynccnt 0x0" ::: "memory");
}

// ------------------------------------------------------------------
// Kernel 1: masked bag-of-words sentence encoding
//   E[bs, d] = sum_l mask[bs,l] * emb[prgrph[bs,l], d];  smask[bs] = any(mask)
// ------------------------------------------------------------------
__global__ void sent_encode_kernel(const int* __restrict__ prgrph,
                                   const unsigned char* __restrict__ wmask,
                                   const float* __restrict__ emb,
                                   float* __restrict__ E,
                                   int* __restrict__ smask) {
  const int bs = blockIdx.x;     // 0..B*S-1
  const int d  = threadIdx.x;    // 0..255
  const int* pr = prgrph + bs * L_;
  const unsigned char* mk = wmask + bs * L_;
  float acc = 0.f;
#pragma unroll
  for (int l = 0; l < L_; ++l) {
    if (mk[l]) acc += emb[(long)pr[l] * D_ + d];
  }
  E[(long)bs * D_ + d] = acc;
  if (d == 0) {
    int any = 0;
#pragma unroll
    for (int l = 0; l < L_; ++l) any |= mk[l];
    smask[bs] = any ? 1 : 0;
  }
}

// ------------------------------------------------------------------
// WMMA fp32 GEMM:  C[M,256] = A[M,256] @ Bm[256,256]
// grid.x = M/16 ; block = 512 (16 waves, wave w owns N-tile n0 = 16*w)
// ------------------------------------------------------------------
__global__ void __launch_bounds__(512)
gemm_wmma_f32_kernel(const float* __restrict__ A,
                     const float* __restrict__ Bm,
                     float* __restrict__ C) {
  __shared__ float As[16 * HS];
  const int tid = threadIdx.x;
  const int m0  = blockIdx.x * 16;

  for (int i = tid; i < 16 * D_; i += 512) {
    const int r = i >> 8, c = i & 255;
    As[r * HS + c] = A[(long)(m0 + r) * D_ + c];
  }
  __syncthreads();

  const int lane = tid & 31;
  const int half = lane >> 4;   // K half-select per WMMA f32 layout
  const int nl   = lane & 15;
  const int n0   = (tid >> 5) * 16;

  v8f acc = {};
#pragma unroll
  for (int kc = 0; kc < 64; ++kc) {
    const int k0 = kc * 4 + 2 * half;
    v2f a, b;
    a[0] = As[nl * HS + k0];
    a[1] = As[nl * HS + k0 + 1];
    b[0] = Bm[(long)k0 * D_ + n0 + nl];
    b[1] = Bm[(long)(k0 + 1) * D_ + n0 + nl];
    acc = __builtin_amdgcn_wmma_f32_16x16x4_f32(false, a, false, b,
                                                (short)0, acc, false, false);
  }
#pragma unroll
  for (int r = 0; r < 8; ++r) {
    C[(long)(m0 + r + 8 * half) * D_ + n0 + nl] = acc[r];
  }
}

// ------------------------------------------------------------------
// Persistent entity-scan kernel. One block owns 16 (b,k) rows and runs all
// 64 steps internally (recurrence is row-independent after hoisting keys@V
// and e@W). grid = B*2 blocks (K padded 30->32), block = 512 threads.
// Wave w computes the 16x16 tile of h@U at columns [16w,16w+16) with the
// U fragments held in registers for the whole kernel. Step inputs e / e@W
// are double-buffered in LDS and prefetched one step ahead with
// GLOBAL_LOAD_ASYNC_TO_LDS_B32 so their L2 latency overlaps compute.
// ------------------------------------------------------------------
__global__ void __launch_bounds__(512, 1)
entity_scan_kernel(const float* __restrict__ E,
                   const float* __restrict__ EW,
                   const float* __restrict__ keys,
                   const float* __restrict__ keysV,
                   const float* __restrict__ U,
                   const int* __restrict__ smask,
                   float* __restrict__ out) {
  __shared__ float Hs [16 * HS];   // current hidden state rows
  __shared__ float Ot [16 * HS];   // candidate update rows
  __shared__ float Ks [16 * HS];   // keys rows (zero for dummy k>=30)
  __shared__ float KVs[16 * HS];   // keys@V rows
  __shared__ float e_s [2][D_];    // double-buffered step inputs
  __shared__ float ew_s[2][D_];
  __shared__ float g_s[16];

  const int tid  = threadIdx.x;
  const int wave = tid >> 5;       // 0..15 -> also "my row" for reductions
  const int lane = tid & 31;
  const int half = lane >> 4;
  const int nl   = lane & 15;
  const int n0   = wave * 16;

  const int b      = blockIdx.x >> 1;
  const int k_base = (blockIdx.x & 1) * 16;

  // ---- kick off async staging of step-0 inputs while we init LDS ----
  {
    const long es0 = (long)b * S_ * D_;
    if (tid < D_) async_ld_f32(E + es0, tid * 4, &e_s[0][tid]);
    else          async_ld_f32(EW + es0, (tid - D_) * 4, &ew_s[0][tid - D_]);
  }

  // ---- init LDS: h0 = 0, load keys / keysV (zero-fill padded rows) ----
  for (int i = tid; i < 16 * HS; i += 512) Hs[i] = 0.f;
  for (int i = tid; i < 16 * D_; i += 512) {
    const int r = i >> 8, c = i & 255;
    const int k = k_base + r;
    float kk = 0.f, kv = 0.f;
    if (k < K_) {
      kk = keys [((long)b * K_ + k) * D_ + c];
      kv = keysV[((long)b * K_ + k) * D_ + c];
    }
    Ks [r * HS + c] = kk;
    KVs[r * HS + c] = kv;
  }

  // ---- preload this wave's 64 B-fragments of U into registers ----
  v2f bf[64];
#pragma unroll
  for (int kc = 0; kc < 64; ++kc) {
    const int k0 = kc * 4 + 2 * half;
    bf[kc][0] = U[(long)k0 * D_ + n0 + nl];
    bf[kc][1] = U[(long)(k0 + 1) * D_ + n0 + nl];
  }
  wait_async0();
  __syncthreads();

  for (int s = 0; s < S_; ++s) {
    const int p = s & 1;
    const int msk = smask[b * S_ + s];          // uniform scalar load, issued early

    // ---- prefetch next step's e / e@W into the other buffer (async) ----
    if (s + 1 < S_) {
      const long es1 = ((long)b * S_ + s + 1) * D_;
      if (tid < D_) async_ld_f32(E + es1, tid * 4, &e_s[p ^ 1][tid]);
      else          async_ld_f32(EW + es1, (tid - D_) * 4, &ew_s[p ^ 1][tid - D_]);
    }

    // ---- gate: g[row] = sigmoid( dot(e, h_row + keys_row) ) ; row = wave ----
    {
      float pr = 0.f;
#pragma unroll
      for (int j = 0; j < 8; ++j) {
        const int d = lane + 32 * j;
        pr += e_s[p][d] * (Hs[wave * HS + d] + Ks[wave * HS + d]);
      }
#pragma unroll
      for (int off = 16; off > 0; off >>= 1) pr += __shfl_xor(pr, off, 32);
      if (lane == 0) g_s[wave] = 1.f / (1.f + __expf(-pr));
    }
    __syncthreads();

    // ---- h@U tile via fp32 WMMA, fused gate+relu+residual ----
    v8f acc = {};
#pragma unroll
    for (int kc = 0; kc < 64; ++kc) {
      const int k0 = kc * 4 + 2 * half;
      v2f a;
      a[0] = Hs[nl * HS + k0];
      a[1] = Hs[nl * HS + k0 + 1];
      acc = __builtin_amdgcn_wmma_f32_16x16x4_f32(false, a, false, bf[kc],
                                                  (short)0, acc, false, false);
    }
#pragma unroll
    for (int r = 0; r < 8; ++r) {
      const int m = r + 8 * half;
      const int n = n0 + nl;
      float ht = acc[r] + KVs[m * HS + n] + ew_s[p][n];  // hU + kV + eW
      ht = fmaxf(ht, 0.f);                               // relu
      Ot[m * HS + n] = Hs[m * HS + n] + g_s[m] * ht;     // gated residual
    }
    __syncthreads();

    // ---- L2-normalize row `wave`; masked update of Hs ----
    {
      float v[8];
      float ss = 0.f;
#pragma unroll
      for (int j = 0; j < 8; ++j) {
        v[j] = Ot[wave * HS + lane + 32 * j];
        ss += v[j] * v[j];
      }
#pragma unroll
      for (int off = 16; off > 0; off >>= 1) ss += __shfl_xor(ss, off, 32);
      if (msk) {
        const float inv = rsqrtf(fmaxf(ss, 1e-12f));
#pragma unroll
        for (int j = 0; j < 8; ++j)
          Hs[wave * HS + lane + 32 * j] = v[j] * inv;
      }
    }
    wait_async0();      // next step's e/ew landed in LDS
    __syncthreads();    // publish Hs + prefetched buffers
  }

  // ---- write final h rows (k < 30 only) ----
  for (int i = tid; i < 16 * D_; i += 512) {
    const int r = i >> 8, c = i & 255;
    const int k = k_base + r;
    if (k < K_) out[((long)b * K_ + k) * D_ + c] = Hs[r * HS + c];
  }
}

// ------------------------------------------------------------------
// Host launcher
// ------------------------------------------------------------------
extern "C" void kernel_launch(void* const* d_in, const int* in_sizes, int n_in,
                              void* d_out, int out_size, void* d_ws, size_t ws_size,
                              hipStream_t stream) {
  (void)in_sizes; (void)n_in; (void)out_size; (void)ws_size;

  const int*           prgrph = (const int*)d_in[0];
  const unsigned char* wmask  = (const unsigned char*)d_in[1];  // jax bool = 1 byte
  const float*         keys   = (const float*)d_in[2];
  const float*         emb    = (const float*)d_in[3];
  const float*         U      = (const float*)d_in[4];
  const float*         V      = (const float*)d_in[5];
  const float*         W      = (const float*)d_in[6];
  float*               out    = (float*)d_out;

  // workspace layout (floats): E[B*S*D] | EW[B*S*D] | keysV[B*K*D] | smask[B*S]
  float* E     = (float*)d_ws;
  float* EW    = E  + (long)B_ * S_ * D_;
  float* keysV = EW + (long)B_ * S_ * D_;
  int*   smask = (int*)(keysV + (long)B_ * K_ * D_);

  // 1) bag-of-words sentence encodings + sentence mask
  sent_encode_kernel<<<B_ * S_, 256, 0, stream>>>(prgrph, wmask, emb, E, smask);

  // 2) hoisted loop-invariant GEMMs (fp32 WMMA)
  gemm_wmma_f32_kernel<<<(B_ * K_) / 16, 512, 0, stream>>>(keys, V, keysV); // keys@V
  gemm_wmma_f32_kernel<<<(B_ * S_) / 16, 512, 0, stream>>>(E,    W, EW);    // e@W (all steps)

  // 3) one persistent kernel runs the whole 64-step recurrence
  entity_scan_kernel<<<B_ * 2, 512, 0, stream>>>(E, EW, keys, keysV, U, smask, out);
}